// GMM_9990093931293
// MI455X (gfx1250) — compile-verified
//
#include <hip/hip_runtime.h>
#include <math.h>

// ---------------------------------------------------------------------------
// GMM energy (DAGMM-style), N=1e6, K=8, D=16, fp32.
// Strategy: lift maha quadratic form to outer-product features so the whole
// per-sample log_prob[k] computation is ONE 16x156x16 fp32 GEMM on the
// matrix pipes (V_WMMA_F32_16X16X4_F32 chained 39x), with k on the M axis so
// logsumexp over k is per-lane (no shuffles).
// ---------------------------------------------------------------------------

typedef float v2f __attribute__((ext_vector_type(2)));
typedef float v8f __attribute__((ext_vector_type(8)));

#define DIM   16
#define NFEAT 156   // 136 quad (d<=e) + 16 linear + 1 const + 3 pad (mult of 4)
#define NSTEP 39    // NFEAT / 4 wmma k-steps
#define WPB   16    // waves per block
#define TPB   (WPB*32)

// ---- compile-time feature index mapping (must match setup kernel order) ----
constexpr int f_type(int j) { return j < 136 ? 0 : (j < 152 ? 1 : (j == 152 ? 2 : 3)); }
constexpr int f_d(int j) {
  if (j >= 136) return 0;
  int d = 0, base = 0;
  while (base + (DIM - d) <= j) { base += DIM - d; ++d; }
  return d;
}
constexpr int f_e(int j) {
  if (j >= 136) return 0;
  int d = f_d(j), base = 0;
  for (int i = 0; i < d; ++i) base += DIM - i;
  return d + (j - base);
}

template <int J>
__device__ __forceinline__ float fval(const float (&zr)[DIM]) {
  if constexpr (f_type(J) == 0)      return zr[f_d(J)] * zr[f_e(J)];
  else if constexpr (f_type(J) == 1) return zr[J - 136];
  else if constexpr (f_type(J) == 2) return 1.0f;
  else                               return 0.0f;
}

// Build the 39 B-operand fragments for this lane's half H (K-slice 4s+2H..+1).
template <int H, int S>
struct Builder {
  static __device__ __forceinline__ void run(v2f (&bf)[NSTEP], const float (&zr)[DIM]) {
    bf[S].x = fval<4 * S + 2 * H>(zr);
    bf[S].y = fval<4 * S + 2 * H + 1>(zr);
    Builder<H, S + 1>::run(bf, zr);
  }
};
template <int H>
struct Builder<H, NSTEP> {
  static __device__ __forceinline__ void run(v2f (&)[NSTEP], const float (&)[DIM]) {}
};

// ---------------------------------------------------------------------------
// Setup: per-k 16x16 Gauss-Jordan inverse + det, symmetrize, fold mu/phi/det
// into the weight matrix W[16][156]; also the scalar penalty. Tiny, one block.
// ---------------------------------------------------------------------------
__global__ void gmm_setup_kernel(const float* __restrict__ phi,
                                 const float* __restrict__ mu,
                                 const float* __restrict__ sigma,
                                 float* __restrict__ Wg,
                                 float* __restrict__ pen_out, int K) {
  int tid = threadIdx.x;
  if (tid == 0) {
    float pen = 0.0f;
    for (int k = 0; k < K; ++k)
      for (int d = 0; d < DIM; ++d)
        pen += 1.0f / (sigma[(k * DIM + d) * DIM + d] + 1e-6f + 1e-12f);
    *pen_out = pen;
  }
  if (tid >= 16) return;
  int row = tid;
  if (row >= K) {  // zero-pad unused output rows
    for (int j = 0; j < NFEAT; ++j) Wg[row * NFEAT + j] = 0.0f;
    return;
  }
  int k = row;
  float a[DIM][DIM], inv[DIM][DIM];
  for (int i = 0; i < DIM; ++i)
    for (int j = 0; j < DIM; ++j) {
      a[i][j]   = sigma[(k * DIM + i) * DIM + j] + (i == j ? 1e-6f : 0.0f);
      inv[i][j] = (i == j) ? 1.0f : 0.0f;
    }
  // Gauss-Jordan with partial pivoting; det = prod(pivots)*sign
  float det = 1.0f;
  for (int c = 0; c < DIM; ++c) {
    int p = c; float best = fabsf(a[c][c]);
    for (int r = c + 1; r < DIM; ++r) {
      float v = fabsf(a[r][c]);
      if (v > best) { best = v; p = r; }
    }
    if (p != c) {
      for (int j = 0; j < DIM; ++j) {
        float t = a[c][j];   a[c][j] = a[p][j];   a[p][j] = t;
        t = inv[c][j]; inv[c][j] = inv[p][j]; inv[p][j] = t;
      }
      det = -det;
    }
    float piv = a[c][c];
    det *= piv;
    float pr = 1.0f / piv;
    for (int j = 0; j < DIM; ++j) { a[c][j] *= pr; inv[c][j] *= pr; }
    for (int r = 0; r < DIM; ++r)
      if (r != c) {
        float f = a[r][c];
        if (f != 0.0f)
          for (int j = 0; j < DIM; ++j) { a[r][j] -= f * a[c][j]; inv[r][j] -= f * inv[c][j]; }
      }
  }
  det = fmaxf(det, 1e-6f);
  float ld = logf(det);
  // symmetric part (quadratic form only sees it) -> reuse a[][]
  for (int i = 0; i < DIM; ++i)
    for (int j = i; j < DIM; ++j) {
      float s = 0.5f * (inv[i][j] + inv[j][i]);
      a[i][j] = s; a[j][i] = s;
    }
  float cvec[DIM]; float dq = 0.0f;
  for (int i = 0; i < DIM; ++i) {
    float s = 0.0f;
    for (int j = 0; j < DIM; ++j) s += a[i][j] * mu[k * DIM + j];
    cvec[i] = s;
  }
  for (int i = 0; i < DIM; ++i) dq += mu[k * DIM + i] * cvec[i];
  // log_softmax(phi)[k]
  float mx = phi[0];
  for (int i = 1; i < K; ++i) mx = fmaxf(mx, phi[i]);
  float se = 0.0f;
  for (int i = 0; i < K; ++i) se += expf(phi[i] - mx);
  float lsm = phi[k] - (mx + logf(se));
  const float log2pi = 1.8378770664093453f;
  float e_k = lsm - 0.5f * ld - 0.5f * (float)DIM * log2pi - 0.5f * dq;
  // write weight row (same feature order as f_d/f_e mapping)
  int j = 0;
  for (int d = 0; d < DIM; ++d)
    for (int e = d; e < DIM; ++e)
      Wg[row * NFEAT + j++] = -(d == e ? 0.5f : 1.0f) * a[d][e];
  for (int d = 0; d < DIM; ++d) Wg[row * NFEAT + j++] = cvec[d];
  Wg[row * NFEAT + j++] = e_k;
  for (; j < NFEAT; ++j) Wg[row * NFEAT + j] = 0.0f;
}

// ---------------------------------------------------------------------------
// Main kernel: one wave handles a 16-sample tile. GEMM: A = W (16 k-rows x 156),
// B = features (156 x 16 samples), D = log_prob (k x sample). Lane n (n<16)
// ends with log_prob[0..7][n] in its 8 accumulator elements -> per-lane LSE.
// ---------------------------------------------------------------------------
__global__ void __launch_bounds__(TPB) gmm_energy_kernel(const float* __restrict__ z,
                                                         const float* __restrict__ Wg,
                                                         float* __restrict__ out,
                                                         int N, int tiles, int K) {
  __shared__ __align__(16) float lds_w[16 * NFEAT];
  const int tid = threadIdx.x;
  for (int i = tid; i < 16 * NFEAT; i += TPB) lds_w[i] = Wg[i];
  __syncthreads();

  const int wave = tid >> 5;
  const int lane = tid & 31;
  const int tile = blockIdx.x * WPB + wave;
  if (tile >= tiles) return;  // wave-uniform exit; EXEC stays full below

  const int n = lane & 15;    // sample column / A-row index
  const int h = lane >> 4;    // wave half -> K-slice offset
  long long row = (long long)tile * 16 + n;
  if (row > (long long)N - 1) row = (long long)N - 1;  // clamp loads for tail

  const float4* zp = (const float4*)(z + row * DIM);
  float4 q0 = zp[0], q1 = zp[1], q2 = zp[2], q3 = zp[3];
  float zr[DIM] = {q0.x, q0.y, q0.z, q0.w, q1.x, q1.y, q1.z, q1.w,
                   q2.x, q2.y, q2.z, q2.w, q3.x, q3.y, q3.z, q3.w};

  // Build this lane's 39 B fragments (divergence fully before any WMMA).
  v2f bf[NSTEP];
  if (h == 0) Builder<0, 0>::run(bf, zr);
  else        Builder<1, 0>::run(bf, zr);

  // A fragments stream from LDS: lane holds W[n][4s+2h .. +1].
  const float* wbase = lds_w + n * NFEAT + 2 * h;  // 8-byte aligned (NFEAT even)

  v8f acc = {0.f, 0.f, 0.f, 0.f, 0.f, 0.f, 0.f, 0.f};
#pragma unroll
  for (int s = 0; s < NSTEP; ++s) {
    v2f afrag = *(const v2f*)(wbase + 4 * s);  // ds_load_b64
    acc = __builtin_amdgcn_wmma_f32_16x16x4_f32(
        false, afrag, false, bf[s], (short)0, acc, false, false);
  }

  // energy[n] = -logsumexp_k(log_prob[k][n]); rows live in acc[0..K-1] (h==0)
  if (h == 0) {
    float mx = acc[0];
#pragma unroll
    for (int v = 1; v < 8; ++v)
      if (v < K) mx = fmaxf(mx, acc[v]);
    float sum = 0.0f;
#pragma unroll
    for (int v = 0; v < 8; ++v)
      if (v < K) sum += __expf(acc[v] - mx);
    float energy = -(mx + __logf(sum));
    long long idx = (long long)tile * 16 + n;
    if (idx < (long long)N) out[idx] = energy;
  }
}

// ---------------------------------------------------------------------------
extern "C" void kernel_launch(void* const* d_in, const int* in_sizes, int n_in,
                              void* d_out, int out_size, void* d_ws, size_t ws_size,
                              hipStream_t stream) {
  const float* z     = (const float*)d_in[0];  // [N,16]
  const float* phi   = (const float*)d_in[1];  // [K]
  const float* mu    = (const float*)d_in[2];  // [K,16]
  const float* sigma = (const float*)d_in[3];  // [K,16,16]

  const int N = in_sizes[0] / DIM;
  const int K = in_sizes[1];
  float* out = (float*)d_out;        // [0..N): energy, [N]: penalty
  float* Wg  = (float*)d_ws;         // 16*156 floats = 9984 B

  gmm_setup_kernel<<<1, 64, 0, stream>>>(phi, mu, sigma, Wg, out + N, K);

  const int tiles  = (N + 15) / 16;
  const int blocks = (tiles + WPB - 1) / WPB;
  gmm_energy_kernel<<<blocks, TPB, 0, stream>>>(z, Wg, out, N, tiles, K);
}